// CollectiveFusionAttention_58677843198309
// MI455X (gfx1250) — compile-verified
//
#include <hip/hip_runtime.h>
#include <cstdint>
#include <math.h>

// Problem constants (match reference)
#define E_          16
#define D_          5
#define B_          8
#define P_          65536
#define EXPERT_DIM_ 128
#define N_BETAS_    4

// ---------------------------------------------------------------------------
// Kernel 1: per-expert scalar weights  w[e] = sigmoid(mean(embed[e])) *
//                                             (1 + mean(sigmoid(betas[e])))
// plus scale = 1/(5*|T|), written to d_ws[0..16].
// ---------------------------------------------------------------------------
__global__ void cf_weights_kernel(const float* __restrict__ betas,
                                  const float* __restrict__ embed,
                                  const float* __restrict__ temperature,
                                  float* __restrict__ ws) {
  const int e = threadIdx.x;
  if (e < E_) {
    float s = 0.f;
#pragma unroll
    for (int i = 0; i < EXPERT_DIM_; ++i) s += embed[e * EXPERT_DIM_ + i];
    const float pos_w = 1.f / (1.f + expf(-(s * (1.f / EXPERT_DIM_))));
    float bs = 0.f;
#pragma unroll
    for (int i = 0; i < N_BETAS_; ++i)
      bs += 1.f / (1.f + expf(-betas[e * N_BETAS_ + i]));
    const float beta_w = 1.f + bs * (1.f / N_BETAS_);
    ws[e] = pos_w * beta_w;
  }
  if (e == 0) ws[E_] = 0.2f / fabsf(temperature[0]);
}

// ---------------------------------------------------------------------------
// Kernel 2: streaming masked weighted reduction.
// Grid: blockIdx.x = tile*8 + b ; 512 blocks x 256 threads.
// Each thread: 4 consecutive patches, one b. V streamed via async
// global->LDS (3-stage pipeline over experts), each lane consumes only the
// 16B/row it copied -> no barriers, per-wave s_wait_asynccnt ordering only.
// ---------------------------------------------------------------------------
__global__ __launch_bounds__(256)
void cf_main_kernel(const float* __restrict__ V,
                    const unsigned char* __restrict__ masks,
                    const float* __restrict__ ws,
                    float* __restrict__ out) {
  __shared__ float lds[3][D_][1024];   // 60 KB triple buffer

  const int tid  = threadIdx.x;
  const int b    = blockIdx.x & (B_ - 1);
  const int tile = blockIdx.x >> 3;
  const int p    = (tile << 10) + (tid << 2);     // base patch of this thread

  // Uniform weight loads (scalar loads)
  float w[E_];
#pragma unroll
  for (int e = 0; e < E_; ++e) w[e] = ws[e];
  const float scale = ws[E_];

  const uint64_t vbase   = (uint64_t)(uintptr_t)V;
  const uint32_t ldsLane =
      (uint32_t)(uintptr_t)(&lds[0][0][0]) + (uint32_t)(tid << 4);

  // Issue one expert-stage: 5 rows (d) of 4KB/block, 16B per lane.
  auto issue = [&](int e, int buf) {
#pragma unroll
    for (int d = 0; d < D_; ++d) {
      const uint32_t ldsOff = ldsLane + (uint32_t)((buf * D_ + d) << 12);
      // byte offset of V[e][d][b][p] : ((e*5+d)*8+b)*P*4 + p*4
      const uint32_t gOff =
          (uint32_t)((((e * D_ + d) * B_ + b) << 18) + (p << 2));
      asm volatile("global_load_async_to_lds_b128 %0, %1, %2"
                   :: "v"(ldsOff), "v"(gOff), "s"(vbase)
                   : "memory");
    }
  };

  float4 acc = make_float4(0.f, 0.f, 0.f, 0.f);
  float4 den = make_float4(0.f, 0.f, 0.f, 0.f);

  issue(0, 0);
  issue(1, 1);

#pragma unroll
  for (int e = 0; e < E_; ++e) {
    if (e + 2 < E_) issue(e + 2, (e + 2) % 3);

    // Guarantee stage e has landed: 5 async ops per later stage in flight.
    if (e < E_ - 2)       asm volatile("s_wait_asynccnt 10" ::: "memory");
    else if (e == E_ - 2) asm volatile("s_wait_asynccnt 5"  ::: "memory");
    else                  asm volatile("s_wait_asynccnt 0"  ::: "memory");

    // 4 mask bytes (jax bool_ = 1 byte) as one aligned u32 load
    const unsigned int mu =
        *(const unsigned int*)(masks + (size_t)e * P_ + p);
    const float we  = w[e];
    const float wm0 = (mu & 0x000000FFu) ? we : 0.f;
    const float wm1 = (mu & 0x0000FF00u) ? we : 0.f;
    const float wm2 = (mu & 0x00FF0000u) ? we : 0.f;
    const float wm3 = (mu & 0xFF000000u) ? we : 0.f;
    den.x += wm0; den.y += wm1; den.z += wm2; den.w += wm3;

    const int buf = e % 3;
#pragma unroll
    for (int d = 0; d < D_; ++d) {
      const float4 v = *(const float4*)&lds[buf][d][tid << 2];
      acc.x = fmaf(wm0, v.x, acc.x);
      acc.y = fmaf(wm1, v.y, acc.y);
      acc.z = fmaf(wm2, v.z, acc.z);
      acc.w = fmaf(wm3, v.w, acc.w);
    }
  }

  const float4 o = make_float4(acc.x * (scale / fmaxf(den.x, 1e-6f)),
                               acc.y * (scale / fmaxf(den.y, 1e-6f)),
                               acc.z * (scale / fmaxf(den.z, 1e-6f)),
                               acc.w * (scale / fmaxf(den.w, 1e-6f)));
  *(float4*)(out + (size_t)b * P_ + p) = o;
}

// ---------------------------------------------------------------------------
// Launch
// Inputs: 0=V (f32), 1=masks (bool, 1B), 2=betas (f32), 3=expert_pos_embed
// (f32), 4=temperature (f32[1]), 5=num_patches (int, unused; P fixed 65536)
// Output: f32 [B=8, P=65536]
// ---------------------------------------------------------------------------
extern "C" void kernel_launch(void* const* d_in, const int* in_sizes, int n_in,
                              void* d_out, int out_size, void* d_ws, size_t ws_size,
                              hipStream_t stream) {
  const float*         V     = (const float*)d_in[0];
  const unsigned char* masks = (const unsigned char*)d_in[1];
  const float*         betas = (const float*)d_in[2];
  const float*         embed = (const float*)d_in[3];
  const float*         temp  = (const float*)d_in[4];
  float*               ws    = (float*)d_ws;   // 17 floats
  float*               out   = (float*)d_out;

  cf_weights_kernel<<<1, 32, 0, stream>>>(betas, embed, temp, ws);

  const int tiles = P_ / 1024;                 // 64
  cf_main_kernel<<<dim3(tiles * B_), dim3(256), 0, stream>>>(V, masks, ws, out);
}